// BrownianKernelLayer_87522843560947
// MI455X (gfx1250) — compile-verified
//
#include <hip/hip_runtime.h>

typedef __attribute__((ext_vector_type(2))) float v2f;
typedef __attribute__((ext_vector_type(8))) float v8f;

#define N_PTS 2048
#define M_PTS 2048
#define DDIM  16

__device__ __forceinline__ float fast_exp2(float x) {
  return __builtin_amdgcn_exp2f(x);       // v_exp_f32 (base-2)
}
__device__ __forceinline__ float fast_log2(float x) {
  return __builtin_amdgcn_logf(x);        // v_log_f32 (base-2)
}

__device__ __forceinline__ float softplus_f(float v) {
  // stable softplus(log_H); log_H ~ 0.29 here but keep it robust
  // exp(v) = exp2(v * log2(e))
  return (v > 20.0f) ? v : log1pf(fast_exp2(v * 1.44269504088896340736f));
}

__device__ __forceinline__ float pow_abs(float a, float p) {
  // |a|^p = exp2(p * log2|a|); a==0 -> exp2(-inf) = 0 (p > 0)
  return fast_exp2(p * fast_log2(fabsf(a)));
}

// t12[0..N) = t1[i] = sum_d |x[i,d]|^p ; t12[N..N+M) = t2[j]
__global__ void fbm_row_terms(const float* __restrict__ x,
                              const float* __restrict__ X2,
                              const float* __restrict__ log_H,
                              float* __restrict__ t12) {
  int t = blockIdx.x * blockDim.x + threadIdx.x;
  float p = 2.0f * softplus_f(log_H[0]);
  const float* row = (t < N_PTS) ? (x + t * DDIM) : (X2 + (t - N_PTS) * DDIM);
  float s = 0.0f;
#pragma unroll
  for (int d = 0; d < DDIM; ++d) s += pow_abs(row[d], p);
  t12[t] = s;
}

// One wave computes one 16x16 output tile.
// Per d: diff_tile = x_col*1^T + 1*(-X2_row)^T via v_wmma_f32_16x16x4_f32,
// then acc += |diff|^p elementwise. Epilogue folds 0.5*(t1+t2) via a second
// WMMA with C = -0.5*acc.
__global__ void __launch_bounds__(256)
fbm_main(const float* __restrict__ x,
         const float* __restrict__ X2,
         const float* __restrict__ log_H,
         const float* __restrict__ t12,
         float* __restrict__ out,
         int use_ws) {
  const int lane = threadIdx.x & 31;
  const int wave = threadIdx.x >> 5;
  const int wid  = blockIdx.x * 8 + wave;      // 0..16383
  const int ti   = wid >> 7;                   // tile row 0..127
  const int tj   = wid & 127;                  // tile col 0..127
  const int ibase = ti * 16;
  const int jbase = tj * 16;
  const int l15   = lane & 15;
  const bool lo   = (lane < 16);

  const float p = 2.0f * softplus_f(log_H[0]);

  // Lanes 0-15 own row (ibase+l15) of x and row (jbase+l15) of X2.
  // Lanes 16-31 mirror the loads (same cachelines) to keep EXEC uniform.
  float xr[DDIM], br[DDIM];
  {
    const float4* xp = (const float4*)(x  + (size_t)(ibase + l15) * DDIM);
    const float4* bp = (const float4*)(X2 + (size_t)(jbase + l15) * DDIM);
#pragma unroll
    for (int q = 0; q < 4; ++q) {
      float4 a = xp[q], b = bp[q];
      xr[4 * q + 0] = a.x; xr[4 * q + 1] = a.y;
      xr[4 * q + 2] = a.z; xr[4 * q + 3] = a.w;
      br[4 * q + 0] = b.x; br[4 * q + 1] = b.y;
      br[4 * q + 2] = b.z; br[4 * q + 3] = b.w;
    }
  }

  const v8f zero8 = {0.f, 0.f, 0.f, 0.f, 0.f, 0.f, 0.f, 0.f};
  v8f acc = zero8;

#pragma unroll
  for (int d = 0; d < DDIM; ++d) {
    // A (16x4 f32): vgpr0 = K0 (lanes<16) / K2 (lanes>=16), vgpr1 = K1 / K3
    // columns: [x[:,d], 1, 0, 0]
    v2f A = { lo ? xr[d] : 0.0f, lo ? 1.0f : 0.0f };
    // B (4x16 f32): vgpr0 = row0 / row2, vgpr1 = row1 / row3
    // rows: [1^T, -X2[:,d]^T, 0, 0]
    v2f B = { lo ? 1.0f : 0.0f, lo ? -br[d] : 0.0f };
    v8f dif = __builtin_amdgcn_wmma_f32_16x16x4_f32(
        false, A, false, B, (short)0, zero8, false, false);
#pragma unroll
    for (int r = 0; r < 8; ++r) acc[r] += pow_abs(dif[r], p);
  }

  // per-point terms
  float t1v, t2v;
  if (use_ws) {
    t1v = t12[ibase + l15];
    t2v = t12[N_PTS + jbase + l15];
  } else {
    t1v = 0.0f; t2v = 0.0f;
#pragma unroll
    for (int d = 0; d < DDIM; ++d) {
      t1v += pow_abs(xr[d], p);
      t2v += pow_abs(br[d], p);
    }
  }

  // K = 0.5*t1[i] + 0.5*t2[j] - 0.5*acc  (rank-2 + C via WMMA)
  v2f A2 = { lo ? 0.5f * t1v : 0.0f, lo ? 1.0f : 0.0f };
  v2f B2 = { lo ? 1.0f : 0.0f, lo ? 0.5f * t2v : 0.0f };
  v8f C;
#pragma unroll
  for (int r = 0; r < 8; ++r) C[r] = -0.5f * acc[r];
  v8f K = __builtin_amdgcn_wmma_f32_16x16x4_f32(
      false, A2, false, B2, (short)0, C, false, false);

  // C-layout: (vgpr r, lane L) -> i = ibase + r + 8*(L>=16), j = jbase + (L&15)
  float* op = out + (size_t)(ibase + (lane >> 4) * 8) * M_PTS + jbase + l15;
#pragma unroll
  for (int r = 0; r < 8; ++r) op[(size_t)r * M_PTS] = K[r];
}

extern "C" void kernel_launch(void* const* d_in, const int* in_sizes, int n_in,
                              void* d_out, int out_size, void* d_ws, size_t ws_size,
                              hipStream_t stream) {
  const float* x     = (const float*)d_in[0];
  const float* X2    = (const float*)d_in[1];
  const float* log_H = (const float*)d_in[2];
  float* out = (float*)d_out;
  float* t12 = (float*)d_ws;

  const int use_ws =
      (ws_size >= (size_t)(N_PTS + M_PTS) * sizeof(float)) ? 1 : 0;

  if (use_ws) {
    fbm_row_terms<<<(N_PTS + M_PTS) / 256, 256, 0, stream>>>(x, X2, log_H, t12);
  }
  // 128x128 tiles, 8 waves (tiles) per 256-thread block
  fbm_main<<<(128 * 128) / 8, 256, 0, stream>>>(x, X2, log_H, t12, out, use_ws);
}